// ContextBlock_90417651516486
// MI455X (gfx1250) — compile-verified
//
#include <hip/hip_runtime.h>
#include <math.h>

// ContextBlock (GCNet 'att' pooling + channel-add MLP), B=16, C=512, L=8192, P=128.
// Memory-bound: 3 full passes over x (256MB each) + 1 full write. MLP done with
// V_WMMA_F32_16X16X4_F32 (exact f32 math, matches reference precision).

#define B_  16
#define C_  512
#define L_  8192
#define P_  128
#define LN_EPS 1e-5f

typedef float v2f __attribute__((ext_vector_type(2)));
typedef float v8f __attribute__((ext_vector_type(8)));

// ---------------- Pass 1: scores[b,l] = sum_c x[b,c,l]*w[c] + bias ----------------
__global__ void __launch_bounds__(256)
scores_kernel(const float* __restrict__ x, const float* __restrict__ w,
              const float* __restrict__ bias, float* __restrict__ scores) {
    const int b = blockIdx.y;
    const int l = blockIdx.x * 256 + threadIdx.x;           // coalesced over l
    const float* xp = x + (size_t)b * C_ * L_ + l;
    float s = 0.0f;
    #pragma unroll 8
    for (int c = 0; c < C_; ++c)
        s = fmaf(xp[(size_t)c * L_], w[c], s);              // w[c] uniform -> scalar load
    scores[b * L_ + l] = s + bias[0];
}

// ---------------- Pass 2: softmax over L per batch ----------------
__global__ void __launch_bounds__(256)
softmax_kernel(const float* __restrict__ scores, float* __restrict__ attn) {
    const int b = blockIdx.x;
    const int tid = threadIdx.x;
    __shared__ float red[256];
    const float* s = scores + (size_t)b * L_;

    float m = -INFINITY;
    for (int l = tid; l < L_; l += 256) m = fmaxf(m, s[l]);
    red[tid] = m; __syncthreads();
    for (int off = 128; off > 0; off >>= 1) {
        if (tid < off) red[tid] = fmaxf(red[tid], red[tid + off]);
        __syncthreads();
    }
    const float mx = red[0];
    __syncthreads();

    float sum = 0.0f;
    for (int l = tid; l < L_; l += 256) sum += expf(s[l] - mx);
    red[tid] = sum; __syncthreads();
    for (int off = 128; off > 0; off >>= 1) {
        if (tid < off) red[tid] += red[tid + off];
        __syncthreads();
    }
    const float inv = 1.0f / red[0];

    float* a = attn + (size_t)b * L_;
    for (int l = tid; l < L_; l += 256) a[l] = expf(s[l] - mx) * inv;
}

// ---------------- Pass 3: context[b,c] = sum_l x[b,c,l]*attn[b,l] ----------------
__global__ void __launch_bounds__(256)
context_kernel(const float* __restrict__ x, const float* __restrict__ attn,
               float* __restrict__ ctx) {
    const int bc = blockIdx.x;                 // 0 .. B*C-1
    const int b = bc >> 9;                     // /C_
    const int tid = threadIdx.x;
    __shared__ float red[256];
    const float* xp = x + (size_t)bc * L_;
    const float* ap = attn + (size_t)b * L_;
    float s = 0.0f;
    #pragma unroll 4
    for (int l = tid; l < L_; l += 256) s = fmaf(xp[l], ap[l], s);
    red[tid] = s; __syncthreads();
    for (int off = 128; off > 0; off >>= 1) {
        if (tid < off) red[tid] += red[tid + off];
        __syncthreads();
    }
    if (tid == 0) ctx[bc] = red[0];
}

// ---------------- Pass 4: MLP via V_WMMA_F32_16X16X4_F32 ----------------
// t = ctx[16x512] @ w1^T + b1 ; LN over P ; ReLU ; add_term = t @ w2^T + b2
__global__ void __launch_bounds__(256)
mlp_wmma_kernel(const float* __restrict__ ctx,
                const float* __restrict__ w1, const float* __restrict__ b1,
                const float* __restrict__ lnw, const float* __restrict__ lnb,
                const float* __restrict__ w2, const float* __restrict__ b2,
                float* __restrict__ add_term) {
    __shared__ float t_lds[16 * P_];           // 8 KB
    const int tid   = threadIdx.x;
    const int wave  = tid >> 5;                // 0..7
    const int lane  = tid & 31;
    const int row   = lane & 15;               // M (batch) for A, N for B/D
    const int khalf = lane >> 4;               // 0: K0/K1, 1: K2/K3 (ISA 16x4 f32 layout)

    // ---- GEMM1: wave handles N-tile = wave (P columns wave*16 .. +15) ----
    {
        v8f acc = {0.f,0.f,0.f,0.f,0.f,0.f,0.f,0.f};
        const int   nglob = wave * 16 + row;
        const float* aptr = ctx + (size_t)row   * C_;   // A[m][k]
        const float* bptr = w1  + (size_t)nglob * C_;   // B[k][n] = w1[n][k]
        for (int k = 0; k < C_; k += 4) {
            v2f a, bb;
            a.x  = aptr[k + 2 * khalf];
            a.y  = aptr[k + 2 * khalf + 1];
            bb.x = bptr[k + 2 * khalf];
            bb.y = bptr[k + 2 * khalf + 1];
            acc = __builtin_amdgcn_wmma_f32_16x16x4_f32(
                    false, a, false, bb, (short)0, acc, false, false);
        }
        const int p = wave * 16 + row;
        #pragma unroll
        for (int v = 0; v < 8; ++v) {
            const int m = v + 8 * khalf;       // D: VGPR v -> M = v (+8 for upper half)
            t_lds[m * P_ + p] = acc[v] + b1[p];
        }
    }
    __syncthreads();

    // ---- LayerNorm(P) + ReLU, one thread per batch row (tiny) ----
    if (tid < 16) {
        float mu = 0.0f;
        for (int p = 0; p < P_; ++p) mu += t_lds[tid * P_ + p];
        mu *= (1.0f / P_);
        float var = 0.0f;
        for (int p = 0; p < P_; ++p) {
            const float d = t_lds[tid * P_ + p] - mu;
            var = fmaf(d, d, var);
        }
        var *= (1.0f / P_);
        const float inv = rsqrtf(var + LN_EPS);
        for (int p = 0; p < P_; ++p) {
            const float v = (t_lds[tid * P_ + p] - mu) * inv * lnw[p] + lnb[p];
            t_lds[tid * P_ + p] = fmaxf(v, 0.0f);
        }
    }
    __syncthreads();

    // ---- GEMM2: add_term[16x512] = t[16x128] @ w2^T; wave owns tiles w, w+8, w+16, w+24 ----
    for (int tile = wave; tile < C_ / 16; tile += 8) {
        v8f acc = {0.f,0.f,0.f,0.f,0.f,0.f,0.f,0.f};
        const int    nglob = tile * 16 + row;
        const float* bptr  = w2 + (size_t)nglob * P_;   // B[k][n] = w2[c][k]
        for (int k = 0; k < P_; k += 4) {
            v2f a, bb;
            a.x  = t_lds[row * P_ + k + 2 * khalf];
            a.y  = t_lds[row * P_ + k + 2 * khalf + 1];
            bb.x = bptr[k + 2 * khalf];
            bb.y = bptr[k + 2 * khalf + 1];
            acc = __builtin_amdgcn_wmma_f32_16x16x4_f32(
                    false, a, false, bb, (short)0, acc, false, false);
        }
        #pragma unroll
        for (int v = 0; v < 8; ++v) {
            const int m = v + 8 * khalf;
            add_term[m * C_ + nglob] = acc[v] + b2[nglob];
        }
    }
}

// ---------------- Pass 5: out = x + add_term[b,c] (float4 streaming) ----------------
__global__ void __launch_bounds__(256)
add_kernel(const float4* __restrict__ x4, const float* __restrict__ add_term,
           float4* __restrict__ out4) {
    const size_t i = (size_t)blockIdx.x * 256 + threadIdx.x;   // float4 index
    const int bc = (int)(i >> 11);                             // / (L_/4)
    const float a = add_term[bc];
    float4 v = x4[i];
    v.x += a; v.y += a; v.z += a; v.w += a;
    out4[i] = v;
}

extern "C" void kernel_launch(void* const* d_in, const int* in_sizes, int n_in,
                              void* d_out, int out_size, void* d_ws, size_t ws_size,
                              hipStream_t stream) {
    const float* x   = (const float*)d_in[0];
    const float* cw  = (const float*)d_in[1];
    const float* cb  = (const float*)d_in[2];
    const float* w1  = (const float*)d_in[3];
    const float* b1  = (const float*)d_in[4];
    const float* lnw = (const float*)d_in[5];
    const float* lnb = (const float*)d_in[6];
    const float* w2  = (const float*)d_in[7];
    const float* b2  = (const float*)d_in[8];
    float* out = (float*)d_out;

    float* ws       = (float*)d_ws;
    float* scores   = ws;                       // B*L = 131072
    float* attn     = ws + (size_t)B_ * L_;     // B*L = 131072
    float* ctx      = attn + (size_t)B_ * L_;   // B*C = 8192
    float* add_term = ctx + (size_t)B_ * C_;    // B*C = 8192

    dim3 g1(L_ / 256, B_);
    scores_kernel<<<g1, 256, 0, stream>>>(x, cw, cb, scores);
    softmax_kernel<<<B_, 256, 0, stream>>>(scores, attn);
    context_kernel<<<B_ * C_, 256, 0, stream>>>(x, attn, ctx);
    mlp_wmma_kernel<<<1, 256, 0, stream>>>(ctx, w1, b1, lnw, lnb, w2, b2, add_term);

    const size_t n4 = (size_t)B_ * C_ * L_ / 4;                // 16,777,216
    add_kernel<<<(unsigned)(n4 / 256), 256, 0, stream>>>(
        (const float4*)x, add_term, (float4*)out);
}